// DQN_7438883356731
// MI455X (gfx1250) — compile-verified
//
#include <hip/hip_runtime.h>

typedef __attribute__((ext_vector_type(2))) float v2f;
typedef __attribute__((ext_vector_type(8))) float v8f;

// ---------------------------------------------------------------------------
// WMMA f32 GEMM: C[M,64] = X[M,K] @ W[K,64] (+ optional bias[64])
// One wave per 16-row strip; 4 column tiles of 16; K consumed 4 at a time via
// V_WMMA_F32_16X16X4_F32.  M must allow wave-uniform tail exit (M%16==0 here).
// ---------------------------------------------------------------------------
__global__ void gemm_xw(const float* __restrict__ X, const float* __restrict__ W,
                        const float* __restrict__ bias, float* __restrict__ C,
                        int M, int K) {
    const int gthread = blockIdx.x * blockDim.x + threadIdx.x;
    const int wave    = gthread >> 5;
    const int lane    = threadIdx.x & 31;
    const int m0      = wave * 16;
    if (m0 >= M) return;                 // wave-uniform (m0 depends only on wave)

    const int half = lane >> 4;          // 0: K offsets {0,1}; 1: K offsets {2,3}
    const int lm   = lane & 15;          // row for A, column for B/C/D

    v8f acc[4];
    #pragma unroll
    for (int nt = 0; nt < 4; ++nt) {
        const float bv = bias ? bias[nt * 16 + lm] : 0.0f;
        #pragma unroll
        for (int v = 0; v < 8; ++v) acc[nt][v] = bv;
    }

    const float* xrow = X + (size_t)(m0 + lm) * K;

    for (int k0 = 0; k0 < K; k0 += 4) {
        const int kb = k0 + 2 * half;
        // A fragment: 16x4 f32, lane layout per ISA 7.12.2
        const float2 av = *(const float2*)(xrow + kb);
        v2f a; a.x = av.x; a.y = av.y;
        #pragma unroll
        for (int nt = 0; nt < 4; ++nt) {
            const int n = nt * 16 + lm;
            v2f b;
            b.x = W[(size_t)kb * 64 + n];
            b.y = W[(size_t)(kb + 1) * 64 + n];
            acc[nt] = __builtin_amdgcn_wmma_f32_16x16x4_f32(
                false, a, false, b, (short)0, acc[nt], false, false);
        }
    }

    // C/D layout: VGPR v -> row v + 8*half, column lm (per tile)
    #pragma unroll
    for (int nt = 0; nt < 4; ++nt) {
        const int n = nt * 16 + lm;
        #pragma unroll
        for (int v = 0; v < 8; ++v) {
            C[(size_t)(m0 + v + 8 * half) * 64 + n] = acc[nt][v];
        }
    }
}

// ---------------------------------------------------------------------------
// Edge scatter for 64-wide features: agg[dst] += rel[etype][src], 16 thr/edge.
// ---------------------------------------------------------------------------
__global__ void edge_scatter64(const float* __restrict__ rel,
                               const int* __restrict__ src,
                               const int* __restrict__ dst,
                               const int* __restrict__ et,
                               float* __restrict__ agg, int E, int N) {
    const long long t = (long long)blockIdx.x * blockDim.x + threadIdx.x;
    const int e = (int)(t >> 4);
    if (e >= E) return;
    const int f = ((int)t & 15) * 4;
    const float4 v = *(const float4*)(rel + ((long long)et[e] * N + src[e]) * 64 + f);
    float* q = agg + (long long)dst[e] * 64 + f;
    atomicAdd(q + 0, v.x);
    atomicAdd(q + 1, v.y);
    atomicAdd(q + 2, v.z);
    atomicAdd(q + 3, v.w);
}

__global__ void relu_k(float* __restrict__ p, int n) {
    const int i = blockIdx.x * blockDim.x + threadIdx.x;
    if (i < n) p[i] = fmaxf(p[i], 0.0f);
}

// ---------------------------------------------------------------------------
// Layer 3: per node, 6 dot products over 64 features (2 rels x 2 + loop x 2).
// ---------------------------------------------------------------------------
__global__ void layer3_node(const float* __restrict__ h2,
                            const float* __restrict__ W3,
                            const float* __restrict__ loop3,
                            const float* __restrict__ b3,
                            float* __restrict__ rel3, float* __restrict__ agg3,
                            int N) {
    const int i = blockIdx.x * blockDim.x + threadIdx.x;
    if (i >= N) return;
    const float* row = h2 + (size_t)i * 64;
    float r00 = 0.f, r01 = 0.f, r10 = 0.f, r11 = 0.f, l0 = 0.f, l1 = 0.f;
    #pragma unroll 8
    for (int k = 0; k < 64; ++k) {
        const float v = row[k];
        r00 += v * W3[k * 2 + 0];
        r01 += v * W3[k * 2 + 1];
        r10 += v * W3[128 + k * 2 + 0];
        r11 += v * W3[128 + k * 2 + 1];
        l0  += v * loop3[k * 2 + 0];
        l1  += v * loop3[k * 2 + 1];
    }
    rel3[(size_t)i * 2 + 0] = r00;
    rel3[(size_t)i * 2 + 1] = r01;
    rel3[(size_t)(N + i) * 2 + 0] = r10;
    rel3[(size_t)(N + i) * 2 + 1] = r11;
    agg3[(size_t)i * 2 + 0] = l0 + b3[0];
    agg3[(size_t)i * 2 + 1] = l1 + b3[1];
}

__global__ void edge_scatter2(const float* __restrict__ rel3,
                              const int* __restrict__ src,
                              const int* __restrict__ dst,
                              const int* __restrict__ et,
                              float* __restrict__ agg3, int E, int N) {
    const int e = blockIdx.x * blockDim.x + threadIdx.x;
    if (e >= E) return;
    const float* p = rel3 + ((long long)et[e] * N + src[e]) * 2;
    float* q = agg3 + (long long)dst[e] * 2;
    atomicAdd(q + 0, p[0]);
    atomicAdd(q + 1, p[1]);
}

// Order-preserving float <-> uint mapping for atomicMin on floats.
__device__ __forceinline__ unsigned f2ord(float f) {
    unsigned u = __float_as_uint(f);
    return (u & 0x80000000u) ? ~u : (u | 0x80000000u);
}
__device__ __forceinline__ float ord2f(unsigned u) {
    return __uint_as_float((u & 0x80000000u) ? (u ^ 0x80000000u) : ~u);
}

__global__ void init_slot(unsigned* slot) { *slot = 0xFFFFFFFFu; }

__global__ void min_reduce(const float* __restrict__ p, int n, unsigned* slot) {
    __shared__ float sm[256];
    float m = 3.402823466e38f;
    for (int i = blockIdx.x * blockDim.x + threadIdx.x; i < n;
         i += gridDim.x * blockDim.x)
        m = fminf(m, p[i]);
    sm[threadIdx.x] = m;
    __syncthreads();
    for (int s = blockDim.x >> 1; s > 0; s >>= 1) {
        if ((int)threadIdx.x < s) sm[threadIdx.x] = fminf(sm[threadIdx.x], sm[threadIdx.x + s]);
        __syncthreads();
    }
    if (threadIdx.x == 0) atomicMin(slot, f2ord(sm[0]));
}

__global__ void finalize_k(const float* __restrict__ agg3,
                           const int* __restrict__ cell,
                           const int* __restrict__ maxs,
                           const unsigned* __restrict__ slot,
                           float* __restrict__ out, int N) {
    const int i = blockIdx.x * blockDim.x + threadIdx.x;
    if (i >= N) return;
    const float gmin = ord2f(*slot) - 1.0f;
    const bool upper = cell[i] >= (maxs[i] - 1);
    const bool lower = cell[i] == 0;
    out[(size_t)i * 2 + 0] = upper ? gmin : agg3[(size_t)i * 2 + 0];
    out[(size_t)i * 2 + 1] = lower ? gmin : agg3[(size_t)i * 2 + 1];
}

extern "C" void kernel_launch(void* const* d_in, const int* in_sizes, int n_in,
                              void* d_out, int out_size, void* d_ws, size_t ws_size,
                              hipStream_t stream) {
    const float* x      = (const float*)d_in[0];
    const int*   src    = (const int*)d_in[1];
    const int*   dst    = (const int*)d_in[2];
    const int*   etypes = (const int*)d_in[3];
    const int*   cell   = (const int*)d_in[4];
    const int*   maxs   = (const int*)d_in[5];
    const float* W1     = (const float*)d_in[6];    // [2,128,64]
    const float* loop1  = (const float*)d_in[7];    // [128,64]
    const float* b1     = (const float*)d_in[8];
    const float* W2     = (const float*)d_in[9];    // [2,64,64]
    const float* loop2  = (const float*)d_in[10];   // [64,64]
    const float* b2     = (const float*)d_in[11];
    const float* W3     = (const float*)d_in[12];   // [2,64,2]
    const float* loop3  = (const float*)d_in[13];   // [64,2]
    const float* b3     = (const float*)d_in[14];
    float* out = (float*)d_out;

    const int N = in_sizes[4];
    const int E = in_sizes[1];
    const int K1 = in_sizes[0] / N;                 // 128

    // Workspace layout (all fully rewritten each call)
    float* relbuf = (float*)d_ws;                   // [2, N, 64]  (reused L1/L2)
    float* h1     = relbuf + (size_t)2 * N * 64;    // [N, 64]
    float* h2     = h1 + (size_t)N * 64;            // [N, 64]
    float* rel3   = h2 + (size_t)N * 64;            // [2, N, 2]
    float* agg3   = rel3 + (size_t)4 * N;           // [N, 2]
    unsigned* slot = (unsigned*)(agg3 + (size_t)2 * N);

    const int waves   = (N + 15) / 16;
    const int gblocks = (waves * 32 + 255) / 256;
    const int eb64    = (int)(((long long)E * 16 + 255) / 256);
    const int nb      = (N + 255) / 256;
    const int fb      = (N * 64 + 255) / 256;
    const int eb2     = (E + 255) / 256;

    // ---- Layer 1 (K=128 -> 64) ----
    gemm_xw<<<gblocks, 256, 0, stream>>>(x, W1,            nullptr, relbuf,                  N, K1);
    gemm_xw<<<gblocks, 256, 0, stream>>>(x, W1 + K1 * 64,  nullptr, relbuf + (size_t)N * 64, N, K1);
    gemm_xw<<<gblocks, 256, 0, stream>>>(x, loop1,         b1,      h1,                      N, K1);
    edge_scatter64<<<eb64, 256, 0, stream>>>(relbuf, src, dst, etypes, h1, E, N);
    relu_k<<<fb, 256, 0, stream>>>(h1, N * 64);

    // ---- Layer 2 (K=64 -> 64) ----
    gemm_xw<<<gblocks, 256, 0, stream>>>(h1, W2,          nullptr, relbuf,                  N, 64);
    gemm_xw<<<gblocks, 256, 0, stream>>>(h1, W2 + 64 * 64, nullptr, relbuf + (size_t)N * 64, N, 64);
    gemm_xw<<<gblocks, 256, 0, stream>>>(h1, loop2,        b2,      h2,                      N, 64);
    edge_scatter64<<<eb64, 256, 0, stream>>>(relbuf, src, dst, etypes, h2, E, N);
    relu_k<<<fb, 256, 0, stream>>>(h2, N * 64);

    // ---- Layer 3 (64 -> 2) + mask epilogue ----
    layer3_node<<<nb, 256, 0, stream>>>(h2, W3, loop3, b3, rel3, agg3, N);
    edge_scatter2<<<eb2, 256, 0, stream>>>(rel3, src, dst, etypes, agg3, E, N);
    init_slot<<<1, 1, 0, stream>>>(slot);
    min_reduce<<<512, 256, 0, stream>>>(agg3, N * 2, slot);
    finalize_k<<<nb, 256, 0, stream>>>(agg3, cell, maxs, slot, out, N);
}